// qInnerAttnM_38886633898202
// MI455X (gfx1250) — compile-verified
//
#include <hip/hip_runtime.h>
#include <hip/hip_bf16.h>
#include <math.h>

#define NF 50
#define THREADS 256
#define BLOCKS_B 1024
#define BLOCKS_C 512

typedef __attribute__((ext_vector_type(2))) float v2f;
typedef __attribute__((ext_vector_type(8))) float v8f;

// Hardware tanh (V_TANH_F32, CDNA5 trans op) instead of ocml polynomial.
__device__ __forceinline__ float htanh(float x) {
#if __has_builtin(__builtin_amdgcn_tanhf)
    return __builtin_amdgcn_tanhf(x);
#elif __has_builtin(__builtin_amdgcn_tanh_f32)
    return __builtin_amdgcn_tanh_f32(x);
#else
    float y;
    asm volatile("v_tanh_f32 %0, %1" : "=v"(y) : "v"(x));
    return y;
#endif
}

// Order-preserving float -> uint encoding (for deterministic atomic max).
__device__ __forceinline__ unsigned enc_f32(float f) {
    unsigned u = __float_as_uint(f);
    return (u & 0x80000000u) ? ~u : (u | 0x80000000u);
}
__device__ __forceinline__ float dec_f32(unsigned u) {
    return __uint_as_float((u & 0x80000000u) ? (u & 0x7fffffffu) : ~u);
}

__global__ void k_init(unsigned* __restrict__ maxu) {
    if (threadIdx.x == 0 && blockIdx.x == 0) *maxu = 0u;  // decodes below any real score
}

// Pass 1: scores[i] = sum_j tanh(qW*q[i][j]) * w[j]; global max via encoded atomicMax.
__global__ void k_scores(const float* __restrict__ q, const float* __restrict__ w,
                         const float* __restrict__ qW, float* __restrict__ scores,
                         unsigned* __restrict__ maxu, int T) {
    __shared__ float smax[THREADS];
    const float a = qW[0];
    const int tid = blockIdx.x * blockDim.x + threadIdx.x;
    const int stride = gridDim.x * blockDim.x;
    float lmax = -INFINITY;
    for (int i = tid; i < T; i += stride) {
        const float* row = q + (size_t)i * NF;
        float s = 0.f;
#pragma unroll
        for (int j = 0; j < NF; j += 2) {
            float2 x = *reinterpret_cast<const float2*>(row + j);  // rows are 8B-aligned (200B stride)
            s += htanh(a * x.x) * w[j];
            s += htanh(a * x.y) * w[j + 1];
        }
        scores[i] = s;
        lmax = fmaxf(lmax, s);
    }
    smax[threadIdx.x] = lmax;
    __syncthreads();
    for (int off = THREADS / 2; off > 0; off >>= 1) {
        if (threadIdx.x < off) smax[threadIdx.x] = fmaxf(smax[threadIdx.x], smax[threadIdx.x + off]);
        __syncthreads();
    }
    if (threadIdx.x == 0) atomicMax(maxu, enc_f32(smax[0]));
}

// Pass 2: z_partial = e^T * Q over this block's rows, via V_WMMA_F32_16X16X4_F32.
// Each wave consumes 4 rows per WMMA step:
//   A (16x4): A[m][k] = e_{r+k}, replicated over M  -> VGPR0 = {e_r | e_{r+2}}, VGPR1 = {e_{r+1} | e_{r+3}}
//   B (4x16): B[k][n] = q[r+k][tile+n]              -> same K split at lane 16
//   D (16x16): all M rows identical; D[0][n] (VGPR0, lanes 0..15) = partial z[tile+n]
__global__ void k_weighted(const float* __restrict__ q, const float* __restrict__ scores,
                           const unsigned* __restrict__ maxu,
                           float* __restrict__ partials, int T) {
    const int WAVES = THREADS / 32;  // 8
    __shared__ float lds_z[WAVES][48];
    __shared__ float lds_t[WAVES][8];
    __shared__ float lds_s[WAVES][32];

    const float M = dec_f32(*maxu);
    const int lane = threadIdx.x & 31;
    const int wave = threadIdx.x >> 5;
    const int gwave = blockIdx.x * WAVES + wave;
    const int nwaves = gridDim.x * WAVES;
    const int groups = T >> 2;  // T = 500000 is divisible by 4

    const int half = lane >> 4;      // 0: K=0/1 rows, 1: K=2/3 rows
    const int col = lane & 15;       // N within tile
    const int tr = (lane >> 1) & 3;  // tail: row offset 0..3 (lanes 8..31 duplicate 0..7)
    const int tc = lane & 1;         // tail: column 48 or 49

    v8f acc0 = {}; v8f acc1 = {}; v8f acc2 = {};
    float acc_s = 0.f;
    float acc_t = 0.f;

    for (int g = gwave; g < groups; g += nwaves) {
        const int r = g << 2;
        const float* row0 = q + (size_t)(r + 2 * half) * NF;
        const float* row1 = row0 + NF;
        const float a0 = __expf(scores[r + 2 * half] - M);
        const float a1 = __expf(scores[r + 2 * half + 1] - M);
        acc_s += a0 + a1;  // per-lane; wave total = 16 * sum(e) for this group

        v2f A; A[0] = a0; A[1] = a1;
        v2f B0, B1, B2;
        B0[0] = row0[col];      B0[1] = row1[col];
        B1[0] = row0[col + 16]; B1[1] = row1[col + 16];
        B2[0] = row0[col + 32]; B2[1] = row1[col + 32];

        acc0 = __builtin_amdgcn_wmma_f32_16x16x4_f32(false, A, false, B0, (short)0, acc0, false, false);
        acc1 = __builtin_amdgcn_wmma_f32_16x16x4_f32(false, A, false, B1, (short)0, acc1, false, false);
        acc2 = __builtin_amdgcn_wmma_f32_16x16x4_f32(false, A, false, B2, (short)0, acc2, false, false);

        // tail columns 48,49 (all lanes compute; only lanes 0..7 are used in the reduction)
        const float et = __expf(scores[r + tr] - M);
        acc_t += et * q[(size_t)(r + tr) * NF + 48 + tc];
    }

    if (lane < 16) {
        lds_z[wave][col]      = acc0[0];
        lds_z[wave][col + 16] = acc1[0];
        lds_z[wave][col + 32] = acc2[0];
    }
    if (lane < 8) lds_t[wave][lane] = acc_t;  // lane = tr*2 + tc
    lds_s[wave][lane] = acc_s;
    __syncthreads();

    const int t = threadIdx.x;
    if (t < 48) {
        float s = 0.f;
        for (int wv = 0; wv < WAVES; ++wv) s += lds_z[wv][t];
        partials[(size_t)blockIdx.x * 51 + t] = s;
    } else if (t < 50) {
        const int c = t - 48;
        float s = 0.f;
        for (int wv = 0; wv < WAVES; ++wv)
            for (int rr = 0; rr < 4; ++rr) s += lds_t[wv][rr * 2 + c];
        partials[(size_t)blockIdx.x * 51 + t] = s;
    } else if (t == 50) {
        float s = 0.f;
        for (int wv = 0; wv < WAVES; ++wv)
            for (int l = 0; l < 32; ++l) s += lds_s[wv][l];
        partials[(size_t)blockIdx.x * 51 + 50] = s * (1.0f / 16.0f);  // undo 16x lane replication
    }
}

// Pass 3: reduce block partials, divide by sum(e), emit z[50].
__global__ void k_final(const float* __restrict__ partials, float* __restrict__ out, int nblocks) {
    __shared__ float stot;
    const int t = threadIdx.x;
    float s = 0.f;
    if (t < 51) {
        for (int b = 0; b < nblocks; ++b) s += partials[(size_t)b * 51 + t];
    }
    if (t == 50) stot = s;
    __syncthreads();
    if (t < 50) out[t] = s / stot;
}

extern "C" void kernel_launch(void* const* d_in, const int* in_sizes, int n_in,
                              void* d_out, int out_size, void* d_ws, size_t ws_size,
                              hipStream_t stream) {
    const float* q    = (const float*)d_in[0];
    const float* ql_w = (const float*)d_in[1];
    const float* qW   = (const float*)d_in[2];
    const int T = in_sizes[0] / NF;

    // d_ws layout (floats): [0..15] maxu + pad, [16 ..) block partials (BLOCKS_C*51), then scores (T)
    unsigned* maxu  = (unsigned*)d_ws;
    float* partials = (float*)d_ws + 16;
    float* scores   = partials + (size_t)BLOCKS_C * 51;

    k_init<<<1, 1, 0, stream>>>(maxu);
    k_scores<<<BLOCKS_B, THREADS, 0, stream>>>(q, ql_w, qW, scores, maxu, T);
    k_weighted<<<BLOCKS_C, THREADS, 0, stream>>>(q, scores, maxu, partials, T);
    k_final<<<1, 64, 0, stream>>>(partials, (float*)d_out, BLOCKS_C);
}